// Decoder_5033701671558
// MI455X (gfx1250) — compile-verified
//
#include <hip/hip_runtime.h>
#include <hip/hip_bf16.h>

// ---------------------------------------------------------------------------
// Decoder (GRU + attention + huge output projection), CDNA5 / gfx1250.
//
// Shapes: V=32000 E=128 H=256 B=64 TY=64 TX=128;  3H=768, E+2H=640, 2H=512.
// Pipeline:
//   K0 init        : zero h0, c0, grid-barrier counter
//   K1 emb gather  : X_emb[t][b][:] = (y==0 ? 0 : emb[y])          (2 MB)
//   K2 hj_proj     : (8192x512)x(512x256) WMMA f32 GEMM            (2.1 GF)
//   K3 recurrence  : persistent, 16 WGs x 512 thr, 3 grid barriers/step,
//                    WMMA f32 16x16x4 for gi/gh with the per-WG weight
//                    slices (172KB) staged in LDS (320KB/WGP on CDNA5),
//                    bank-conflict-free via +4-float row pitch.
//   K4 out proj    : (4096x768)x(768x64000) WMMA f32 GEMM, fused
//                    pairwise max + bias.  Grid ordered m-fastest so W_mo
//                    (196.6MB, ~L2-sized) streams from HBM exactly once:
//                    ~733MB total -> ~31us at 23.3 TB/s.  global_prefetch
//                    keeps the W_mo stream 3 cachelines ahead.
// All matrix math uses V_WMMA_F32_16X16X4_F32 (full fp32; the big GEMM is
// HBM-bound so lower-precision WMMA would not be faster).
// ---------------------------------------------------------------------------

typedef __attribute__((ext_vector_type(2))) float v2f;
typedef __attribute__((ext_vector_type(8))) float v8f;

#define NWG 16          // workgroups in persistent recurrence kernel
#define WIH_PITCH 644   // 640 + 4  (pitch % 64 == 4 -> conflict-free LDS)
#define WHH_PITCH 260   // 256 + 4

__device__ __forceinline__ v8f wmma4(v2f a, v2f b, v8f c) {
  // D = A(16x4) * B(4x16) + C(16x16), fp32
  return __builtin_amdgcn_wmma_f32_16x16x4_f32(
      /*neg_a=*/false, a, /*neg_b=*/false, b,
      /*c_mod=*/(short)0, c, /*reuse_a=*/false, /*reuse_b=*/false);
}

__device__ __forceinline__ float sigmoid_f(float x) {
  return 1.0f / (1.0f + __expf(-x));
}
__device__ __forceinline__ float tanh_fast(float x) {
  float ax = fabsf(x);
  float e  = __expf(-2.0f * ax);        // in (0,1], no overflow
  float t  = (1.0f - e) / (1.0f + e);
  return copysignf(t, x);
}

// Software grid barrier: monotonically increasing counter + epoch target.
__device__ __forceinline__ void grid_sync(unsigned* bar, unsigned& epoch) {
  epoch += NWG;
  __syncthreads();
  if (threadIdx.x == 0) {
    __threadfence();                    // release prior global writes
    atomicAdd(bar, 1u);
    while (__hip_atomic_load(bar, __ATOMIC_RELAXED, __HIP_MEMORY_SCOPE_AGENT)
           < epoch) {
      __builtin_amdgcn_s_sleep(1);
    }
  }
  __syncthreads();
  __threadfence();                      // acquire peer writes
}

// ---------------------------------------------------------------------------
// K0: zero h0 (64x256), c0 (64x512), barrier counter
__global__ void init_kernel(float* h_cur, float* x_c, unsigned* bar) {
  int i = blockIdx.x * 256 + threadIdx.x;
  if (i < 64 * 256) h_cur[i] = 0.0f;
  if (i < 64 * 512) x_c[i] = 0.0f;
  if (i == 0) *bar = 0u;
}

// ---------------------------------------------------------------------------
// K1: X_emb[(t*64+b)*128 + e] = emb_eff[y[b][t]][e]   (row 0 of emb zeroed)
__global__ void embgather_kernel(const int* __restrict__ y,
                                 const float* __restrict__ emb,
                                 float* __restrict__ Xemb) {
  int blk = blockIdx.x;           // t*64 + b
  int t = blk >> 6, b = blk & 63;
  int e = threadIdx.x;            // 0..127
  int idx = y[b * 64 + t];
  float v = (idx == 0) ? 0.0f : emb[(size_t)idx * 128 + e];
  Xemb[(size_t)blk * 128 + e] = v;
}

// ---------------------------------------------------------------------------
// K2: hj_proj[(b*128+tx)*256 + h] = sum_k h_j[b][tx][k] * W_a[h][256+k]
// M=8192, N=256, K=512.  grid(16,64) x 256thr; one 16x16 tile per wave.
__global__ __launch_bounds__(256) void hjproj_kernel(
    const float* __restrict__ h_j, const float* __restrict__ W_a,
    float* __restrict__ hjp) {
  int wave = threadIdx.x >> 5, lane = threadIdx.x & 31;
  int ln16 = lane & 15, half = lane >> 4;
  int m_tile = blockIdx.y * 8 + wave;   // 0..511
  int n0 = blockIdx.x * 16;             // 0..255 step 16
  const float* a_base = h_j + (size_t)(m_tile * 16 + ln16) * 512;
  const float* b_base = W_a + (size_t)(n0 + ln16) * 768 + 256;
  v8f acc = {};
#pragma unroll 4
  for (int j = 0; j < 128; ++j) {
    int col = 4 * j + 2 * half;
    v2f a; a[0] = a_base[col]; a[1] = a_base[col + 1];
    v2f b; b[0] = b_base[col]; b[1] = b_base[col + 1];
    acc = wmma4(a, b, acc);
  }
#pragma unroll
  for (int r = 0; r < 8; ++r) {
    int m = m_tile * 16 + r + 8 * half;
    hjp[(size_t)m * 256 + n0 + ln16] = acc[r];
  }
}

// ---------------------------------------------------------------------------
// K3: persistent recurrence. 16 WGs x 512 threads (16 waves).
//   Startup : stage this WG's 48-row slices of W_ih (48x640) and W_hh
//             (48x256) into LDS with padded pitch (conflict-free b64 reads).
//   Phase A : gi = [Xemb_t | c] @ W_ih^T + b_ih   (K=640)
//             gh = h @ W_hh^T + b_hh              (K=256)   -> 24 WMMA tiles
//   Phase B : GRU gates -> h (=s_t), store into S[t][b][0:256]
//   Phase C : per-b attention: s_proj, energy(tanh), softmax, c_new;
//             store c into x_c and S[t][b][256:768]
__global__ __launch_bounds__(512) void recurrence_kernel(
    const float* __restrict__ Xemb, const float* __restrict__ hjp,
    const float* __restrict__ h_j,  const float* __restrict__ W_ih,
    const float* __restrict__ W_hh, const float* __restrict__ b_ih,
    const float* __restrict__ b_hh, const float* __restrict__ W_a,
    const float* __restrict__ v_a,
    float* __restrict__ gi, float* __restrict__ gh,
    float* __restrict__ h_cur, float* __restrict__ x_c,
    float* __restrict__ Sbuf, unsigned* bar) {
  const int wg = blockIdx.x;
  const int tid = threadIdx.x;
  const int wave = tid >> 5;
  const int lane = tid & 31;
  const int ln16 = lane & 15;
  const int half = lane >> 4;
  unsigned epoch = 0;

  __shared__ float sh_s[2][256];
  __shared__ float sh_sproj[2][256];
  __shared__ float sh_red[2][128];
  __shared__ float sh_alpha[2][128];
  extern __shared__ float sh_dyn[];            // 48*WIH_PITCH + 48*WHH_PITCH
  float* sh_wih = sh_dyn;                      // [48][WIH_PITCH]
  float* sh_whh = sh_dyn + 48 * WIH_PITCH;     // [48][WHH_PITCH]

  // ---- stage weight slices into LDS (once) --------------------------------
  for (int idx = tid; idx < 48 * 640; idx += 512) {
    int r = idx / 640, k = idx - r * 640;
    sh_wih[r * WIH_PITCH + k] = W_ih[(size_t)(wg * 48 + r) * 640 + k];
  }
  for (int idx = tid; idx < 48 * 256; idx += 512) {
    int r = idx >> 8, k = idx & 255;
    sh_whh[r * WHH_PITCH + k] = W_hh[(size_t)(wg * 48 + r) * 256 + k];
  }
  __syncthreads();

  for (int t = 0; t < 64; ++t) {
    // ---------------- Phase A: gi (tasks 0..11) and gh (tasks 12..23) ------
    for (int task = wave; task < 24; task += 16) {
      int isGh = (task >= 12);
      int sub = isGh ? task - 12 : task;
      int mt = sub & 3;               // 4 M-tiles (rows 0..63)
      int nt = sub >> 2;              // 3 N-tiles within this WG's 48 cols
      int nl = nt * 16 + ln16;        // local row in LDS weight slab
      int n0 = wg * 48 + nt * 16;
      int row = mt * 16 + ln16;
      int nn = n0 + ln16;
      v8f acc;
      float bias = isGh ? b_hh[nn] : b_ih[nn];
#pragma unroll
      for (int r = 0; r < 8; ++r) acc[r] = bias;

      if (!isGh) {
        const float* Xt = Xemb + (size_t)t * 64 * 128;
        const float* wr = sh_wih + nl * WIH_PITCH;
#pragma unroll 4
        for (int j = 0; j < 32; ++j) {          // K = 0..127 from embedding
          int col = 4 * j + 2 * half;
          v2f a; a[0] = Xt[row * 128 + col]; a[1] = Xt[row * 128 + col + 1];
          v2f b; b[0] = wr[col]; b[1] = wr[col + 1];
          acc = wmma4(a, b, acc);
        }
#pragma unroll 4
        for (int j = 0; j < 128; ++j) {         // K = 128..639 from c_{t-1}
          int col = 4 * j + 2 * half;
          v2f a; a[0] = x_c[row * 512 + col]; a[1] = x_c[row * 512 + col + 1];
          v2f b; b[0] = wr[128 + col]; b[1] = wr[128 + col + 1];
          acc = wmma4(a, b, acc);
        }
#pragma unroll
        for (int r = 0; r < 8; ++r) {
          int m = mt * 16 + r + 8 * half;
          gi[m * 768 + nn] = acc[r];
        }
      } else {
        const float* wr = sh_whh + nl * WHH_PITCH;
#pragma unroll 4
        for (int j = 0; j < 64; ++j) {          // K = 0..255 from h_{t-1}
          int col = 4 * j + 2 * half;
          v2f a; a[0] = h_cur[row * 256 + col]; a[1] = h_cur[row * 256 + col + 1];
          v2f b; b[0] = wr[col]; b[1] = wr[col + 1];
          acc = wmma4(a, b, acc);
        }
#pragma unroll
        for (int r = 0; r < 8; ++r) {
          int m = mt * 16 + r + 8 * half;
          gh[m * 768 + nn] = acc[r];
        }
      }
    }
    grid_sync(bar, epoch);

    // ---------------- Phase B: gates, h update -----------------------------
    for (int e = wg * 512 + tid; e < 64 * 256; e += NWG * 512) {
      int b = e >> 8, h = e & 255;
      float ir = gi[b * 768 + h];
      float iz = gi[b * 768 + 256 + h];
      float in_ = gi[b * 768 + 512 + h];
      float hr = gh[b * 768 + h];
      float hz = gh[b * 768 + 256 + h];
      float hn = gh[b * 768 + 512 + h];
      float r = sigmoid_f(ir + hr);
      float z = sigmoid_f(iz + hz);
      float n = tanh_fast(in_ + r * hn);
      float hp = h_cur[b * 256 + h];
      float s = (1.0f - z) * n + z * hp;
      h_cur[b * 256 + h] = s;
      Sbuf[(size_t)(t * 64 + b) * 768 + h] = s;
    }
    grid_sync(bar, epoch);

    // ---------------- Phase C: attention (4 b per WG, 2 at a time) ---------
    for (int rnd = 0; rnd < 2; ++rnd) {
      int bb = tid >> 8;                  // 0/1: which b of the pair
      int b = wg * 4 + rnd * 2 + bb;
      int lt = tid & 255;
      sh_s[bb][lt] = h_cur[b * 256 + lt];
      __syncthreads();
      // s_proj[h] = sum_k s[k] * W_a[h][k]
      {
        float acc = 0.0f;
        const float* wr = W_a + (size_t)lt * 768;
        for (int k = 0; k < 256; ++k) acc += sh_s[bb][k] * wr[k];
        sh_sproj[bb][lt] = acc;
      }
      __syncthreads();
      // energy[tx] = sum_h tanh(s_proj[h] + hjp[b][tx][h]) * v_a[h]
      int tx = lt & 127, seg = lt >> 7;
      {
        const float* hp = hjp + (size_t)(b * 128 + tx) * 256 + seg * 128;
        const float* sp = &sh_sproj[bb][seg * 128];
        const float* va = v_a + seg * 128;
        float acc = 0.0f;
        for (int h = 0; h < 128; ++h) acc += tanh_fast(sp[h] + hp[h]) * va[h];
        if (seg) sh_red[bb][tx] = acc;
        __syncthreads();
        if (!seg) sh_alpha[bb][tx] = acc + sh_red[bb][tx];   // energies
        __syncthreads();
      }
      // softmax over 128
      {
        if (!seg) sh_red[bb][tx] = sh_alpha[bb][tx];
        __syncthreads();
        for (int s = 64; s > 0; s >>= 1) {
          if (!seg && tx < s)
            sh_red[bb][tx] = fmaxf(sh_red[bb][tx], sh_red[bb][tx + s]);
          __syncthreads();
        }
        float emax = sh_red[bb][0];
        __syncthreads();
        if (!seg) sh_alpha[bb][tx] = __expf(sh_alpha[bb][tx] - emax);
        __syncthreads();
        if (!seg) sh_red[bb][tx] = sh_alpha[bb][tx];
        __syncthreads();
        for (int s = 64; s > 0; s >>= 1) {
          if (!seg && tx < s) sh_red[bb][tx] += sh_red[bb][tx + s];
          __syncthreads();
        }
        float inv = 1.0f / sh_red[bb][0];
        __syncthreads();
        if (!seg) sh_alpha[bb][tx] *= inv;
        __syncthreads();
      }
      // c_new[k] = sum_tx alpha[tx] * h_j[b][tx][k]
      for (int kk = lt; kk < 512; kk += 256) {
        float acc = 0.0f;
        const float* hjr = h_j + (size_t)b * 128 * 512 + kk;
        for (int x = 0; x < 128; ++x) acc += sh_alpha[bb][x] * hjr[x * 512];
        x_c[b * 512 + kk] = acc;
        Sbuf[(size_t)(t * 64 + b) * 768 + 256 + kk] = acc;
      }
      __syncthreads();
    }
    grid_sync(bar, epoch);
  }
}

// ---------------------------------------------------------------------------
// K4: out[(b*64+t)*V + v] = max_p( b_mo[p*V+v] + sum_k S[t][b][k]*W_mo[p*V+v][k] )
// M=4096 (rows t*64+b), 32000 v's, K=768.  grid(32,2000) x 256thr:
// blockIdx.x (m-group) varies fastest so all 256 m-tiles consume a W_mo slab
// while it is L2-resident -> W_mo streams from HBM exactly once.
// One (16m x 16v) double-tile (p=0 and p=1) per wave; max fused at store.
__global__ __launch_bounds__(256) void outproj_kernel(
    const float* __restrict__ Sbuf, const float* __restrict__ W_mo,
    const float* __restrict__ b_mo, float* __restrict__ out) {
  int wave = threadIdx.x >> 5, lane = threadIdx.x & 31;
  int ln16 = lane & 15, half = lane >> 4;
  int m_tile = blockIdx.x * 8 + wave;     // 0..255
  int v0 = blockIdx.y * 16;               // 0..31984 step 16
  int nn = v0 + ln16;
  const float* a_base = Sbuf + (size_t)(m_tile * 16 + ln16) * 768;
  const float* b0_base = W_mo + (size_t)nn * 768;
  const float* b1_base = W_mo + (size_t)(32000 + nn) * 768;
  v8f acc0, acc1;
  float bz0 = b_mo[nn], bz1 = b_mo[32000 + nn];
#pragma unroll
  for (int r = 0; r < 8; ++r) { acc0[r] = bz0; acc1[r] = bz1; }
  const int ja = 2 * half;
  for (int jc = 0; jc < 24; ++jc) {       // 24 chunks x 8 K-steps (128B/lane)
    if (jc < 21) {                        // stream W_mo 3 cachelines ahead
      __builtin_prefetch(b0_base + (jc + 3) * 32 + ja, 0, 0);
      __builtin_prefetch(b1_base + (jc + 3) * 32 + ja, 0, 0);
    }
#pragma unroll
    for (int jj = 0; jj < 8; ++jj) {
      int col = jc * 32 + jj * 4 + ja;
      v2f a;  a[0]  = a_base[col];  a[1]  = a_base[col + 1];
      v2f b0; b0[0] = b0_base[col]; b0[1] = b0_base[col + 1];
      v2f b1; b1[0] = b1_base[col]; b1[1] = b1_base[col + 1];
      acc0 = wmma4(a, b0, acc0);
      acc1 = wmma4(a, b1, acc1);
    }
  }
#pragma unroll
  for (int r = 0; r < 8; ++r) {
    int m = m_tile * 16 + r + 8 * half;
    int t = m >> 6, b = m & 63;
    out[(size_t)(b * 64 + t) * 32000 + nn] = fmaxf(acc0[r], acc1[r]);
  }
}

// ---------------------------------------------------------------------------
extern "C" void kernel_launch(void* const* d_in, const int* in_sizes, int n_in,
                              void* d_out, int out_size, void* d_ws,
                              size_t ws_size, hipStream_t stream) {
  (void)in_sizes; (void)n_in; (void)out_size; (void)ws_size;
  const int*   y    = (const int*)d_in[0];
  const float* h_j  = (const float*)d_in[1];
  const float* emb  = (const float*)d_in[2];
  const float* W_ih = (const float*)d_in[3];
  const float* W_hh = (const float*)d_in[4];
  const float* b_ih = (const float*)d_in[5];
  const float* b_hh = (const float*)d_in[6];
  const float* W_a  = (const float*)d_in[7];
  const float* v_a  = (const float*)d_in[8];
  const float* W_mo = (const float*)d_in[9];
  const float* b_mo = (const float*)d_in[10];
  float* out = (float*)d_out;

  float* ws = (float*)d_ws;
  unsigned* bar = (unsigned*)ws;                    // 16 floats reserved
  float* hjp   = ws + 16;                           // 64*128*256
  float* Xemb  = hjp + 64 * 128 * 256;              // 64*64*128
  float* Sbuf  = Xemb + 64 * 64 * 128;              // 64*64*768
  float* gi    = Sbuf + 64 * 64 * 768;              // 64*768
  float* gh    = gi + 64 * 768;                     // 64*768
  float* h_cur = gh + 64 * 768;                     // 64*256
  float* x_c   = h_cur + 64 * 256;                  // 64*512
  // total ~5.92M floats (~23.7 MB)

  size_t rec_lds = (size_t)(48 * WIH_PITCH + 48 * WHH_PITCH) * sizeof(float);

  init_kernel<<<128, 256, 0, stream>>>(h_cur, x_c, bar);
  embgather_kernel<<<64 * 64, 128, 0, stream>>>(y, emb, Xemb);
  hjproj_kernel<<<dim3(16, 64), 256, 0, stream>>>(h_j, W_a, hjp);
  recurrence_kernel<<<NWG, 512, rec_lds, stream>>>(Xemb, hjp, h_j, W_ih, W_hh,
                                                   b_ih, b_hh, W_a, v_a, gi,
                                                   gh, h_cur, x_c, Sbuf, bar);
  outproj_kernel<<<dim3(32, 2000), 256, 0, stream>>>(Sbuf, W_mo, b_mo, out);
}